// GNNLayer_188978561195
// MI455X (gfx1250) — compile-verified
//
#include <hip/hip_runtime.h>

typedef __attribute__((ext_vector_type(2))) float v2f;
typedef __attribute__((ext_vector_type(8))) float v8f;

#define GNN_N_NODES 100000
#define GNN_D_IN    256
#define GNN_D_OUT   64
#define GNN_N_EDGES 1600000

// ---------------------------------------------------------------------------
// Kernel 1: support = features @ W   (f32 WMMA 16x16x4)
// Block: 256 threads (8 waves). Block tile: 64 rows x 64 cols.
// Wave w: N-tile = w&3 (16 cols), M-tiles at rowbase = blk*64 + (w>>2)*32.
// Two independent accumulator chains per wave hide WMMA latency; the B
// fragment is loaded once per k-step and reused by both WMMAs.
// ---------------------------------------------------------------------------
__global__ __launch_bounds__(256) void gnn_gemm_wmma(
    const float* __restrict__ A,   // [N_NODES, D_IN]
    const float* __restrict__ W,   // [D_IN, D_OUT]
    float* __restrict__ S)         // [N_NODES, D_OUT]
{
    const int lane   = threadIdx.x & 31;
    const int wave   = threadIdx.x >> 5;
    const int ntile  = wave & 3;          // which 16-col tile
    const int mgroup = wave >> 2;         // 0 or 1
    const int half   = lane >> 4;         // half-wave: 0 -> K {0,1}, 1 -> K {2,3}
    const int l16    = lane & 15;
    const int koff   = half * 2;
    const int n0     = ntile * 16;

    const int rowbase = blockIdx.x * 64 + mgroup * 32;

    // A-fragment source rows for the two M-tiles (clamped, never branch:
    // EXEC must stay all-ones for WMMA)
    int r0c = min(rowbase + l16,      GNN_N_NODES - 1);
    int r1c = min(rowbase + 16 + l16, GNN_N_NODES - 1);

    const float* a0p = A + (long)r0c * GNN_D_IN;
    const float* a1p = A + (long)r1c * GNN_D_IN;
    const float* wp  = W + n0 + l16;

    v8f acc0 = {};
    v8f acc1 = {};

    #pragma unroll 4
    for (int k = 0; k < GNN_D_IN; k += 4) {
        // A 16x4: lane holds A[m][k+koff], A[m][k+koff+1]  (contiguous b64 load)
        v2f af0 = *(const v2f*)(a0p + k + koff);
        v2f af1 = *(const v2f*)(a1p + k + koff);
        // B 4x16: lane holds W[k+koff][n], W[k+koff+1][n]  (WGP$-resident)
        v2f bf;
        bf.x = wp[(k + koff) * GNN_D_OUT];
        bf.y = wp[(k + koff + 1) * GNN_D_OUT];
        acc0 = __builtin_amdgcn_wmma_f32_16x16x4_f32(
            false, af0, false, bf, (short)0, acc0, false, false);
        acc1 = __builtin_amdgcn_wmma_f32_16x16x4_f32(
            false, af1, false, bf, (short)0, acc1, false, false);
    }

    // C/D layout: VGPR v -> M = v + 8*half, N = n0 + l16.
    // Wave-uniform fast path: all 32 rows of this wave in range -> 16
    // unpredicated stores with constant offsets (no exec manipulation).
    if (rowbase + 32 <= GNN_N_NODES) {
        float* sp = S + (long)(rowbase + half * 8) * GNN_D_OUT + n0 + l16;
        #pragma unroll
        for (int v = 0; v < 8; ++v) {
            sp[(long)v * GNN_D_OUT]        = acc0[v];   // rows rowbase+half*8+v
            sp[(long)(v + 16) * GNN_D_OUT] = acc1[v];   // rows +16
        }
    } else {
        #pragma unroll
        for (int v = 0; v < 8; ++v) {
            int m = v + half * 8;
            int row0 = rowbase + m;
            int row1 = rowbase + 16 + m;
            if (row0 < GNN_N_NODES) S[(long)row0 * GNN_D_OUT + n0 + l16] = acc0[v];
            if (row1 < GNN_N_NODES) S[(long)row1 * GNN_D_OUT + n0 + l16] = acc1[v];
        }
    }
}

// ---------------------------------------------------------------------------
// Kernel 2: zero the output accumulator (d_out is poisoned by the harness)
// ---------------------------------------------------------------------------
__global__ __launch_bounds__(256) void gnn_zero(float4* __restrict__ out, int n4)
{
    int i = blockIdx.x * blockDim.x + threadIdx.x;
    if (i < n4) out[i] = make_float4(0.f, 0.f, 0.f, 0.f);
}

// ---------------------------------------------------------------------------
// Kernel 3: edge-parallel SpMM: out[r] += v * support[c]
// One wave per edge (tid>>5 uniform per wave); each lane does 2 columns.
// support/out are L2-resident (25.6 MB each), so gathers + f32 atomics hit L2.
// ---------------------------------------------------------------------------
__global__ __launch_bounds__(256) void gnn_spmm(
    const int* __restrict__ rows,
    const int* __restrict__ cols,
    const float* __restrict__ vals,
    const float* __restrict__ S,
    float* __restrict__ out)
{
    long tid = (long)blockIdx.x * blockDim.x + threadIdx.x;
    if (tid >= (long)GNN_N_EDGES * 32) return;
    int e  = (int)(tid >> 5);
    int c2 = ((int)tid & 31) * 2;

    int   r = rows[e];
    int   c = cols[e];
    float v = vals[e];

    const float* sp = S + (long)c * GNN_D_OUT + c2;
    float*       op = out + (long)r * GNN_D_OUT + c2;

    v2f sv = *(const v2f*)sp;
    atomicAdd(op,     v * sv.x);
    atomicAdd(op + 1, v * sv.y);
}

// ---------------------------------------------------------------------------
// Kernel 4: in-place ReLU on the aggregated output
// ---------------------------------------------------------------------------
__global__ __launch_bounds__(256) void gnn_relu(float4* __restrict__ out, int n4)
{
    int i = blockIdx.x * blockDim.x + threadIdx.x;
    if (i < n4) {
        float4 x = out[i];
        x.x = fmaxf(x.x, 0.f);
        x.y = fmaxf(x.y, 0.f);
        x.z = fmaxf(x.z, 0.f);
        x.w = fmaxf(x.w, 0.f);
        out[i] = x;
    }
}

extern "C" void kernel_launch(void* const* d_in, const int* in_sizes, int n_in,
                              void* d_out, int out_size, void* d_ws, size_t ws_size,
                              hipStream_t stream)
{
    const float* features = (const float*)d_in[0];   // [100000, 256]
    const float* weight   = (const float*)d_in[1];   // [256, 64]
    const int*   adj_rows = (const int*)d_in[2];     // [1600000]
    const int*   adj_cols = (const int*)d_in[3];     // [1600000]
    const float* adj_vals = (const float*)d_in[4];   // [1600000]
    float*       out      = (float*)d_out;           // [100000, 64]
    float*       support  = (float*)d_ws;            // [100000, 64] scratch

    // 1) support = features @ W  (WMMA)
    {
        int grid = (GNN_N_NODES + 63) / 64;
        gnn_gemm_wmma<<<grid, 256, 0, stream>>>(features, weight, support);
    }

    // 2) out = 0
    {
        int n4 = (GNN_N_NODES * GNN_D_OUT) / 4;
        int grid = (n4 + 255) / 256;
        gnn_zero<<<grid, 256, 0, stream>>>((float4*)out, n4);
    }

    // 3) out[r] += v * support[c]  (edge-parallel, f32 atomics)
    {
        long threads = (long)GNN_N_EDGES * 32;
        int grid = (int)((threads + 255) / 256);
        gnn_spmm<<<grid, 256, 0, stream>>>(adj_rows, adj_cols, adj_vals, support, out);
    }

    // 4) out = relu(out)
    {
        int n4 = (GNN_N_NODES * GNN_D_OUT) / 4;
        int grid = (n4 + 255) / 256;
        gnn_relu<<<grid, 256, 0, stream>>>((float4*)out, n4);
    }
}